// TokenReduction_41729902248203
// MI455X (gfx1250) — compile-verified
//
#include <hip/hip_runtime.h>
#include <hip/hip_bf16.h>
#include <math.h>

// Fixed problem shape from the reference setup_inputs()
#define NB   32
#define TT   1024
#define CC   768
#define RR   256
#define DD   (TT - RR)   // 768

typedef __attribute__((ext_vector_type(2))) float v2f;
typedef __attribute__((ext_vector_type(8))) float v8f;

__device__ __forceinline__ v8f wmma_f32_k4(v2f a, v2f b, v8f c) {
  // D(16x16,f32) = A(16x4,f32) * B(4x16,f32) + C
  return __builtin_amdgcn_wmma_f32_16x16x4_f32(false, a, false, b, (short)0, c,
                                               false, false);
}

// Batched 16-deep K step: issue 4 A loads + 4 B loads, then 4 WMMAs.
__device__ __forceinline__ v8f wmma_k16(const float* __restrict__ arow,
                                        const float* __restrict__ brow,
                                        float invj, int k, v8f acc) {
  v2f a0 = *(const v2f*)(arow + k);
  v2f a1 = *(const v2f*)(arow + k + 4);
  v2f a2 = *(const v2f*)(arow + k + 8);
  v2f a3 = *(const v2f*)(arow + k + 12);
  v2f b0 = *(const v2f*)(brow + k);
  v2f b1 = *(const v2f*)(brow + k + 4);
  v2f b2 = *(const v2f*)(brow + k + 8);
  v2f b3 = *(const v2f*)(brow + k + 12);
  b0 *= invj; b1 *= invj; b2 *= invj; b3 *= invj;
  acc = wmma_f32_k4(a0, b0, acc);
  acc = wmma_f32_k4(a1, b1, acc);
  acc = wmma_f32_k4(a2, b2, acc);
  acc = wmma_f32_k4(a3, b3, acc);
  return acc;
}

// Async global->LDS copy of one 16-byte chunk (per lane). LDS target address is
// the low 32 bits of the generic shared pointer (flat LDS aperture truncation).
__device__ __forceinline__ void async_copy_b128(float* lds_dst, const float* gsrc) {
  unsigned lds_addr = (unsigned)(size_t)lds_dst;
  asm volatile("global_load_async_to_lds_b128 %0, %1, off"
               :: "v"(lds_addr), "v"(gsrc) : "memory");
}
__device__ __forceinline__ void async_wait0() {
  asm volatile("s_wait_asynccnt 0x0" ::: "memory");
}

// ---------------------------------------------------------------------------
// 1) inv_norm[row] = 1/||metric[row,:]||  (one wave per row)
__global__ __launch_bounds__(256) void k_rownorm(const float* __restrict__ metric,
                                                 float* __restrict__ inv_norm) {
  int w = threadIdx.x >> 5, lane = threadIdx.x & 31;
  int row = blockIdx.x * 8 + w;
  const float* p = metric + (size_t)row * CC;
  float s = 0.f;
  for (int j = lane; j < CC; j += 32) { float v = p[j]; s += v * v; }
  for (int m = 16; m > 0; m >>= 1) s += __shfl_xor(s, m, 32);
  if (lane == 0) inv_norm[row] = 1.0f / sqrtf(s);
}

// 2) qn[b,:] = normalize(query[b, T-1, :])  (one wave per batch)
__global__ __launch_bounds__(256) void k_qnorm(const float* __restrict__ query,
                                               float* __restrict__ qn) {
  int w = threadIdx.x >> 5, lane = threadIdx.x & 31;
  int b = blockIdx.x * 8 + w;
  const float* q = query + ((size_t)b * TT + (TT - 1)) * CC;
  float s = 0.f;
  for (int j = lane; j < CC; j += 32) { float v = q[j]; s += v * v; }
  for (int m = 16; m > 0; m >>= 1) s += __shfl_xor(s, m, 32);
  float rinv = 1.0f / sqrtf(s);
  for (int j = lane; j < CC; j += 32) qn[(size_t)b * CC + j] = q[j] * rinv;
}

// 3) imp[row] = inv_norm[row] * dot(metric[row,:], qn[b,:])  (one wave per row)
__global__ __launch_bounds__(256) void k_imp(const float* __restrict__ metric,
                                             const float* __restrict__ inv_norm,
                                             const float* __restrict__ qn,
                                             float* __restrict__ imp) {
  int w = threadIdx.x >> 5, lane = threadIdx.x & 31;
  int row = blockIdx.x * 8 + w;
  int b = row >> 10;
  const float* p = metric + (size_t)row * CC;
  const float* q = qn + (size_t)b * CC;
  float s = 0.f;
  for (int j = lane; j < CC; j += 32) s += p[j] * q[j];
  for (int m = 16; m > 0; m >>= 1) s += __shfl_xor(s, m, 32);
  if (lane == 0) imp[row] = s * inv_norm[row];
}

// ---------------------------------------------------------------------------
// 4/6) streaming sim = m m^T via WMMA f32 16x16x4.
// mode 0: rowmax of masked sim.  mode 1: masked-max restricted to rank[j] > rank[i].
__global__ __launch_bounds__(256) void k_simmax(const float* __restrict__ metric,
                                                const float* __restrict__ inv_norm,
                                                const int* __restrict__ rank,
                                                float* __restrict__ outMax,
                                                int mode) {
  __shared__ __align__(16) float lA[16 * CC];   // normalized A panel
  __shared__ float lred[8][16];
  __shared__ int lrank[16];
  int b = blockIdx.x >> 6;
  int rt = blockIdx.x & 63;
  int r0 = rt * 16;
  int tid = threadIdx.x;

  // async stage raw A panel (16 contiguous rows = 48KB) into LDS
  {
    const float* gsrc = metric + ((size_t)b * TT + r0) * CC;
#pragma unroll
    for (int it = 0; it < (16 * CC) / (4 * 256); ++it) {   // 12 issues/thread
      int chunk = tid + it * 256;
      async_copy_b128(lA + chunk * 4, gsrc + chunk * 4);
    }
    async_wait0();
  }
  if (mode && tid < 16) lrank[tid] = rank[b * TT + r0 + tid];
  __syncthreads();
  // normalize in place
  for (int idx = tid; idx < 16 * CC; idx += 256) {
    int rr = idx / CC;
    lA[idx] *= inv_norm[b * TT + r0 + rr];
  }
  __syncthreads();

  int w = tid >> 5, lane = tid & 31;
  int lrow = lane & 15, lhalf = lane >> 4;
  float rmax[8];
#pragma unroll
  for (int g = 0; g < 8; ++g) rmax[g] = -INFINITY;

  for (int ct = w; ct < 64; ct += 8) {
    int col = ct * 16 + lrow;
    float invj = inv_norm[b * TT + col];
    int rkj = mode ? rank[b * TT + col] : 0;
    const float* arow = lA + lrow * CC + 2 * lhalf;
    const float* brow = metric + ((size_t)b * TT + col) * CC + 2 * lhalf;
    v8f acc = {};
    for (int k = 0; k < CC; k += 16) {
      __builtin_prefetch(brow + k + 256, 0, 1);  // speculative, no guard
      acc = wmma_k16(arow, brow, invj, k, acc);
    }
    bool jprot = (col == 0) || (col == TT - 1);
#pragma unroll
    for (int g = 0; g < 8; ++g) {
      int i = r0 + g + 8 * lhalf;
      float vv = acc[g];
      bool dead = jprot || (i == col) || (i == 0) || (i == TT - 1);
      if (mode) dead = dead || (rkj <= lrank[g + 8 * lhalf]);
      if (!dead && vv > rmax[g]) rmax[g] = vv;
    }
  }
#pragma unroll
  for (int g = 0; g < 8; ++g) {
    float v = rmax[g];
    for (int m = 8; m > 0; m >>= 1) v = fmaxf(v, __shfl_xor(v, m, 32));
    rmax[g] = v;
  }
  if (lrow == 0)
#pragma unroll
    for (int g = 0; g < 8; ++g) lred[w][8 * lhalf + g] = rmax[g];
  __syncthreads();
  if (tid < 16) {
    float v = -INFINITY;
    for (int ww = 0; ww < 8; ++ww) v = fmaxf(v, lred[ww][tid]);
    outMax[(size_t)b * TT + r0 + tid] = v;
  }
}

// 5/8) per-batch bitonic sort of 1024 (key, idx) pairs.
// desc=1: order by (key desc, idx asc)    [order = argsort(-key), stable]
// desc=0: order by (key asc, idx asc)     [argsort(key), stable]
__global__ __launch_bounds__(256) void k_sort(const float* __restrict__ keys,
                                              int* __restrict__ order_out,
                                              int* __restrict__ rank_out, int desc) {
  __shared__ float sv[TT];
  __shared__ int si[TT];
  int b = blockIdx.x, tid = threadIdx.x;
  for (int i = tid; i < TT; i += 256) { sv[i] = keys[(size_t)b * TT + i]; si[i] = i; }
  __syncthreads();
  for (int k = 2; k <= TT; k <<= 1) {
    for (int j = k >> 1; j > 0; j >>= 1) {
      for (int i = tid; i < TT; i += 256) {
        int ixj = i ^ j;
        if (ixj > i) {
          float av = sv[i], bv = sv[ixj];
          int ai = si[i], bi = si[ixj];
          bool before = desc ? (av > bv || (av == bv && ai < bi))
                             : (av < bv || (av == bv && ai < bi));
          bool up = ((i & k) == 0);
          if (up ? !before : before) {
            sv[i] = bv; sv[ixj] = av; si[i] = bi; si[ixj] = ai;
          }
        }
      }
      __syncthreads();
    }
  }
  for (int i = tid; i < TT; i += 256) {
    if (order_out) order_out[(size_t)b * TT + i] = si[i];
    if (rank_out) rank_out[(size_t)b * TT + si[i]] = i;
  }
}

// 7) scores = imp - maskedmax
__global__ __launch_bounds__(256) void k_scores(const float* __restrict__ imp,
                                                const float* __restrict__ mmax,
                                                float* __restrict__ scores) {
  int i = blockIdx.x * 256 + threadIdx.x;
  scores[i] = imp[i] - mmax[i];
}

// 9) dst_idx[s] = argmax_d sim[src_s, dst_d]  (gathered-row WMMA GEMM)
__global__ __launch_bounds__(256) void k_dstidx(const float* __restrict__ metric,
                                               const float* __restrict__ inv_norm,
                                               const int* __restrict__ edge,
                                               int* __restrict__ dst_idx) {
  __shared__ __align__(16) float lA[16 * CC];
  __shared__ int stok[16];
  __shared__ float lrv[8][16];
  __shared__ int lri[8][16];
  int b = blockIdx.x >> 4;
  int st = blockIdx.x & 15;
  int s0 = st * 16;
  int tid = threadIdx.x;
  if (tid < 16) stok[tid] = edge[(size_t)b * TT + s0 + tid];
  __syncthreads();
  // async stage gathered A rows into LDS (raw), then normalize in place
  {
#pragma unroll
    for (int it = 0; it < (16 * CC) / (4 * 256); ++it) {   // 12 issues/thread
      int chunk = tid + it * 256;
      int rr = chunk / (CC / 4);
      int kk = chunk - rr * (CC / 4);
      const float* gsrc = metric + ((size_t)b * TT + stok[rr]) * CC + kk * 4;
      async_copy_b128(lA + chunk * 4, gsrc);
    }
    async_wait0();
  }
  __syncthreads();
  for (int idx = tid; idx < 16 * CC; idx += 256) {
    int rr = idx / CC;
    lA[idx] *= inv_norm[b * TT + stok[rr]];
  }
  __syncthreads();

  int w = tid >> 5, lane = tid & 31;
  int lrow = lane & 15, lhalf = lane >> 4;
  float bval[8]; int bidx[8];
#pragma unroll
  for (int g = 0; g < 8; ++g) { bval[g] = -INFINITY; bidx[g] = 0x7fffffff; }

  for (int ct = w; ct < 48; ct += 8) {
    int dpos = ct * 16 + lrow;
    int dtok = edge[(size_t)b * TT + RR + dpos];
    float invj = inv_norm[b * TT + dtok];
    const float* arow = lA + lrow * CC + 2 * lhalf;
    const float* brow = metric + ((size_t)b * TT + dtok) * CC + 2 * lhalf;
    v8f acc = {};
    for (int k = 0; k < CC; k += 16) {
      __builtin_prefetch(brow + k + 256, 0, 1);
      acc = wmma_k16(arow, brow, invj, k, acc);
    }
    bool jprot = (dtok == 0) || (dtok == TT - 1);
#pragma unroll
    for (int g = 0; g < 8; ++g) {
      float vv = jprot ? -INFINITY : acc[g];
      if (vv > bval[g] || (vv == bval[g] && dpos < bidx[g])) { bval[g] = vv; bidx[g] = dpos; }
    }
  }
#pragma unroll
  for (int g = 0; g < 8; ++g) {
    float v = bval[g]; int i = bidx[g];
    for (int m = 8; m > 0; m >>= 1) {
      float ov = __shfl_xor(v, m, 32); int oi = __shfl_xor(i, m, 32);
      if (ov > v || (ov == v && oi < i)) { v = ov; i = oi; }
    }
    bval[g] = v; bidx[g] = i;
  }
  if (lrow == 0)
#pragma unroll
    for (int g = 0; g < 8; ++g) { lrv[w][8 * lhalf + g] = bval[g]; lri[w][8 * lhalf + g] = bidx[g]; }
  __syncthreads();
  if (tid < 16) {
    float v = -INFINITY; int i = 0x7fffffff;
    for (int ww = 0; ww < 8; ++ww) {
      float ov = lrv[ww][tid]; int oi = lri[ww][tid];
      if (ov > v || (ov == v && oi < i)) { v = ov; i = oi; }
    }
    dst_idx[(size_t)b * RR + s0 + tid] = i;
  }
}

// 10) per-src merge coefficients: a = 2*softmax_src, bcoef = 2*softmax_dst - 1
__global__ __launch_bounds__(256) void k_coef(const float* __restrict__ imp,
                                              const int* __restrict__ edge,
                                              const int* __restrict__ dst_idx,
                                              float* __restrict__ aArr,
                                              float* __restrict__ bArr) {
  int b = blockIdx.x, s = threadIdx.x;
  int d = dst_idx[(size_t)b * RR + s];
  int stok = edge[(size_t)b * TT + s];
  int dtok = edge[(size_t)b * TT + RR + d];
  float ws = imp[(size_t)b * TT + stok];
  float wd = imp[(size_t)b * TT + dtok];
  float mx = fmaxf(ws, wd);
  float e1 = expf(ws - mx), e2 = expf(wd - mx);
  float inv = 1.0f / (e1 + e2);
  aArr[(size_t)b * RR + s] = 2.0f * e1 * inv;
  bArr[(size_t)b * RR + s] = 2.0f * e2 * inv - 1.0f;
}

// 11) output-row mapping: p-th output row <-> p-th smallest dst token
__global__ __launch_bounds__(256) void k_outpos(const int* __restrict__ edge,
                                                int* __restrict__ out_tok,
                                                int* __restrict__ out_d) {
  __shared__ int flag[TT];
  __shared__ int posOf[TT];
  __shared__ int tsum[256];
  int b = blockIdx.x, tid = threadIdx.x;
  for (int i = tid; i < TT; i += 256) flag[i] = 0;
  __syncthreads();
  flag[edge[(size_t)b * TT + tid]] = 1;                   // RR == 256 src marks
  for (int i = tid; i < DD; i += 256) posOf[edge[(size_t)b * TT + RR + i]] = i;
  __syncthreads();
  int base = tid * 4, cnt = 0, lc[4];
#pragma unroll
  for (int q = 0; q < 4; ++q) { lc[q] = cnt; cnt += (flag[base + q] == 0); }
  tsum[tid] = cnt;
  __syncthreads();
  for (int off = 1; off < 256; off <<= 1) {
    int v = (tid >= off) ? tsum[tid - off] : 0;
    __syncthreads();
    tsum[tid] += v;
    __syncthreads();
  }
  int excl = tsum[tid] - cnt;
#pragma unroll
  for (int q = 0; q < 4; ++q) {
    int v = base + q;
    if (!flag[v]) {
      int p = excl + lc[q];
      out_tok[(size_t)b * DD + p] = v;
      out_d[(size_t)b * DD + p] = posOf[v];
    }
  }
}

// 12) final merged output row: out[p,:] = merged / ts_new
__global__ __launch_bounds__(256) void k_final(const float* __restrict__ x,
                                               const float* __restrict__ tsz,
                                               const int* __restrict__ edge,
                                               const int* __restrict__ dst_idx,
                                               const float* __restrict__ aArr,
                                               const float* __restrict__ bArr,
                                               const int* __restrict__ out_tok,
                                               const int* __restrict__ out_d,
                                               float* __restrict__ out) {
  __shared__ int mtok[RR];
  __shared__ float mcoef[RR];
  __shared__ int s_v, s_cnt;
  __shared__ float s_scale, s_denom, s_tsv;
  int b = blockIdx.x / DD, p = blockIdx.x % DD;
  int tid = threadIdx.x;
  if (tid == 0) {
    int d = out_d[(size_t)b * DD + p];
    int v = out_tok[(size_t)b * DD + p];
    float bs = 0.f, tss = 0.f;
    int cnt = 0;
    for (int s = 0; s < RR; ++s) {                 // serial: deterministic order
      if (dst_idx[(size_t)b * RR + s] == d) {
        int stok = edge[(size_t)b * TT + s];
        float tsrc = tsz[(size_t)b * TT + stok];
        mtok[cnt] = stok;
        mcoef[cnt] = aArr[(size_t)b * RR + s] * tsrc;
        bs += bArr[(size_t)b * RR + s];
        tss += tsrc;
        ++cnt;
      }
    }
    float tsv = tsz[(size_t)b * TT + v];
    s_v = v; s_cnt = cnt; s_tsv = tsv;
    s_scale = 1.0f + bs;
    s_denom = tsv + tss;
  }
  __syncthreads();
  int v = s_v, cnt = s_cnt;
  float tsv = s_tsv, scale = s_scale, rden = 1.0f / s_denom;
  const float* xr = x + ((size_t)b * TT + v) * CC;
  for (int ci = tid; ci < CC; ci += 256) {
    float acc = xr[ci] * tsv * scale;
    for (int m = 0; m < cnt; ++m)
      acc += mcoef[m] * x[((size_t)b * TT + mtok[m]) * CC + ci];
    out[((size_t)b * DD + p) * CC + ci] = acc * rden;
  }
}

// ---------------------------------------------------------------------------
extern "C" void kernel_launch(void* const* d_in, const int* in_sizes, int n_in,
                              void* d_out, int out_size, void* d_ws, size_t ws_size,
                              hipStream_t stream) {
  (void)in_sizes; (void)n_in; (void)out_size; (void)ws_size;
  const float* x      = (const float*)d_in[0];
  const float* query  = (const float*)d_in[1];
  const float* metric = (const float*)d_in[2];
  const float* tsz    = (const float*)d_in[3];
  float* out = (float*)d_out;

  float* wsf = (float*)d_ws;
  float* inv_norm = wsf;                       // 32768
  float* qn       = inv_norm + NB * TT;        // 24576
  float* imp      = qn + NB * CC;              // 32768
  float* rowmax   = imp + NB * TT;             // 32768
  float* mmax     = rowmax + NB * TT;          // 32768
  float* scores   = mmax + NB * TT;            // 32768
  float* aArr     = scores + NB * TT;          // 8192
  float* bArr     = aArr + NB * RR;            // 8192
  int* rank    = (int*)(bArr + NB * RR);       // 32768
  int* edge    = rank + NB * TT;               // 32768
  int* dstI    = edge + NB * TT;               // 8192
  int* out_tok = dstI + NB * RR;               // 24576
  int* out_d   = out_tok + NB * DD;            // 24576

  k_rownorm<<<NB * TT / 8, 256, 0, stream>>>(metric, inv_norm);
  k_qnorm<<<NB / 8, 256, 0, stream>>>(query, qn);
  k_imp<<<NB * TT / 8, 256, 0, stream>>>(metric, inv_norm, qn, imp);
  k_simmax<<<NB * (TT / 16), 256, 0, stream>>>(metric, inv_norm, rank, rowmax, 0);
  k_sort<<<NB, 256, 0, stream>>>(rowmax, nullptr, rank, 1);
  k_simmax<<<NB * (TT / 16), 256, 0, stream>>>(metric, inv_norm, rank, mmax, 1);
  k_scores<<<NB * TT / 256, 256, 0, stream>>>(imp, mmax, scores);
  k_sort<<<NB, 256, 0, stream>>>(scores, edge, nullptr, 0);
  k_dstidx<<<NB * (RR / 16), 256, 0, stream>>>(metric, inv_norm, edge, dstI);
  k_coef<<<NB, 256, 0, stream>>>(imp, edge, dstI, aArr, bArr);
  k_outpos<<<NB, 256, 0, stream>>>(edge, out_tok, out_d);
  k_final<<<NB * DD, 256, 0, stream>>>(x, tsz, edge, dstI, aArr, bArr,
                                       out_tok, out_d, out);
}